// AttentionLayer_6459630813575
// MI455X (gfx1250) — compile-verified
//
#include <hip/hip_runtime.h>
#include <math.h>

#define D     1024
#define NROWS 50000

typedef __attribute__((ext_vector_type(2))) float v2f;
typedef __attribute__((ext_vector_type(8))) float v8f;

// workspace layout (float offsets)
#define WS_BW    0      // b_w_in [1024], b_w_out [1024]
#define WS_ACC   2048   // acc_in [1024], acc_out [1024]  (softmax numerators)
#define WS_ASUM  4096   // asum_in, asum_out
#define WS_RES   5120   // res_in [1024], res_out [1024]  (contiguous -> attn[2048])

// ---------------- kernel 0: zero the cross-block accumulators ----------------
__global__ void zero_accum(float* __restrict__ ws) {
    int i = blockIdx.x * blockDim.x + threadIdx.x;
    if (i < 2050) ws[WS_ACC + i] = 0.0f;   // acc_in, acc_out, asum_in, asum_out
}

// ---------------- kernel 1: b_w = b_alpha^T @ W_alpha (per module) ----------------
__global__ __launch_bounds__(256) void precompute_bw(
        const float* __restrict__ Wa_in,  const float* __restrict__ ba_in,
        const float* __restrict__ Wa_out, const float* __restrict__ ba_out,
        float* __restrict__ ws) {
    int m = blockIdx.x >> 2;                       // module: 0=in, 1=out
    const float* W = m ? Wa_out : Wa_in;
    const float* b = m ? ba_out : ba_in;
    int d = (blockIdx.x & 3) * 256 + threadIdx.x;  // 4 blocks x 256 = 1024 cols
    float acc = 0.0f;
    for (int k = 0; k < D; ++k)
        acc = fmaf(b[k], W[(size_t)k * D + d], acc);   // coalesced over d
    ws[WS_BW + m * D + d] = acc;
}

// ---------------- kernel 2: single-pass softmax-weighted row sum over X ----------------
// wave-per-row; lane owns 32 columns (8 float4 chunks, coalesced); accumulator in VGPRs
__global__ __launch_bounds__(256) void stream_softmax(
        const float* __restrict__ Xin, const float* __restrict__ Xout,
        float* __restrict__ ws) {
    const int m    = blockIdx.y;
    const float* X = m ? Xout : Xin;
    const float* bw = ws + WS_BW + m * D;
    float* accg  = ws + WS_ACC + m * D;
    float* asumg = ws + WS_ASUM + m;

    const int lane = threadIdx.x & 31;
    const int wave = threadIdx.x >> 5;             // 8 waves / block

    const float4* bw4 = (const float4*)bw;
    float4 bwr[8];
    #pragma unroll
    for (int i = 0; i < 8; ++i) bwr[i] = bw4[i * 32 + lane];

    float4 acc[8];
    #pragma unroll
    for (int i = 0; i < 8; ++i) acc[i] = make_float4(0.f, 0.f, 0.f, 0.f);
    float asum = 0.0f;

    for (int row = blockIdx.x * 8 + wave; row < NROWS; row += gridDim.x * 8) {
        const float4* xr = (const float4*)(X + (size_t)row * D);
        float4 x[8];
        float p = 0.0f;
        #pragma unroll
        for (int i = 0; i < 8; ++i) {
            x[i] = xr[i * 32 + lane];              // wave reads 512B contiguous per i
            p = fmaf(x[i].x, bwr[i].x, p);
            p = fmaf(x[i].y, bwr[i].y, p);
            p = fmaf(x[i].z, bwr[i].z, p);
            p = fmaf(x[i].w, bwr[i].w, p);
        }
        #pragma unroll
        for (int off = 16; off > 0; off >>= 1)
            p += __shfl_xor(p, off, 32);           // all lanes get full dot
        float w = expf(p);                         // e^{a_w.X0} cancels in normalization
        asum += w;
        #pragma unroll
        for (int i = 0; i < 8; ++i) {
            acc[i].x = fmaf(w, x[i].x, acc[i].x);
            acc[i].y = fmaf(w, x[i].y, acc[i].y);
            acc[i].z = fmaf(w, x[i].z, acc[i].z);
            acc[i].w = fmaf(w, x[i].w, acc[i].w);
        }
    }

    // block-level reduction: each wave dumps to its own LDS slice, then tree-sum
    __shared__ float sacc[8][D];
    __shared__ float ssum[8];
    #pragma unroll
    for (int i = 0; i < 8; ++i) {
        int base = (i * 32 + lane) * 4;
        sacc[wave][base + 0] = acc[i].x;
        sacc[wave][base + 1] = acc[i].y;
        sacc[wave][base + 2] = acc[i].z;
        sacc[wave][base + 3] = acc[i].w;
    }
    if (lane == 0) ssum[wave] = asum;
    __syncthreads();

    for (int c = threadIdx.x; c < D; c += 256) {
        float s = 0.0f;
        #pragma unroll
        for (int wv = 0; wv < 8; ++wv) s += sacc[wv][c];
        atomicAdd(&accg[c], s);
    }
    if (threadIdx.x == 0) {
        float s = 0.0f;
        #pragma unroll
        for (int wv = 0; wv < 8; ++wv) s += ssum[wv];
        atomicAdd(asumg, s);
    }
}

// ---------------- kernel 3: res = (acc/sum_alpha) @ W_sum via fp32 WMMA ----------------
// one wave per 16-column tile; A = vector padded to 16 rows (row 0 live), K stepped by 4
__global__ __launch_bounds__(32) void wsum_matvec_wmma(
        const float* __restrict__ Wsum_in, const float* __restrict__ Wsum_out,
        float* __restrict__ ws) {
    const int m    = blockIdx.x >> 6;              // 64 tiles per module
    const int tile = blockIdx.x & 63;
    const float* W = m ? Wsum_out : Wsum_in;
    const float* s = ws + WS_ACC + m * D;
    const float inv = 1.0f / ws[WS_ASUM + m];

    const int lane  = threadIdx.x;                 // 0..31, full wave (EXEC all-1s)
    const int half  = (lane >> 4) & 1;             // lanes 16-31 carry K+2,K+3
    const int nlane = lane & 15;
    const int koff  = half * 2;
    const bool aact = (nlane == 0);                // only lanes 0 and 16 carry M=0 of A
    const int d0    = tile * 16;

    v8f c = {};
    for (int k = 0; k < D; k += 4) {
        float s0 = s[k + koff + 0];
        float s1 = s[k + koff + 1];
        v2f a, b;
        a.x = aact ? s0 : 0.0f;                    // A[M=0][K=k+koff]
        a.y = aact ? s1 : 0.0f;                    // A[M=0][K=k+koff+1]
        b.x = W[(size_t)(k + koff + 0) * D + d0 + nlane];   // B[K][N=nlane]
        b.y = W[(size_t)(k + koff + 1) * D + d0 + nlane];
        c = __builtin_amdgcn_wmma_f32_16x16x4_f32(
                /*neg_a=*/false, a, /*neg_b=*/false, b,
                /*c_mod=*/(short)0, c, /*reuse_a=*/false, /*reuse_b=*/false);
    }
    // D row M=0 lives in VGPR0 of lanes 0-15 (N = lane)
    if (lane < 16) ws[WS_RES + m * D + d0 + lane] = c[0] * inv;
}

// ---------------- kernel 4: out = elu(attn @ lin_W^T + lin_b) ----------------
__global__ __launch_bounds__(256) void final_linear(
        const float* __restrict__ lin_W, const float* __restrict__ lin_b,
        const float* __restrict__ ws, float* __restrict__ out) {
    const int wave = threadIdx.x >> 5;
    const int lane = threadIdx.x & 31;
    const int i = blockIdx.x * 8 + wave;           // 128 blocks x 8 waves = 1024 outputs
    const float* attn = ws + WS_RES;               // [2048] = res_in ++ res_out
    const float* row  = lin_W + (size_t)i * (2 * D);
    float p = 0.0f;
    for (int j = lane; j < 2 * D; j += 32)
        p = fmaf(attn[j], row[j], p);              // coalesced along the row
    #pragma unroll
    for (int off = 16; off > 0; off >>= 1)
        p += __shfl_xor(p, off, 32);
    if (lane == 0) {
        float v = p + lin_b[i];
        out[i] = (v > 0.0f) ? v : expm1f(v);       // ELU(alpha=1)
    }
}

extern "C" void kernel_launch(void* const* d_in, const int* in_sizes, int n_in,
                              void* d_out, int out_size, void* d_ws, size_t ws_size,
                              hipStream_t stream) {
    const float* X_in   = (const float*)d_in[0];
    const float* X_out  = (const float*)d_in[1];
    const float* Wa_in  = (const float*)d_in[2];
    // d_in[3] = a_alpha_in  : unused (e^{a_w.X0} cancels in the normalized sum)
    const float* ba_in  = (const float*)d_in[4];
    const float* Ws_in  = (const float*)d_in[5];
    const float* Wa_out = (const float*)d_in[6];
    // d_in[7] = a_alpha_out : unused
    const float* ba_out = (const float*)d_in[8];
    const float* Ws_out = (const float*)d_in[9];
    const float* lin_W  = (const float*)d_in[10];
    const float* lin_b  = (const float*)d_in[11];
    float* ws  = (float*)d_ws;
    float* out = (float*)d_out;

    zero_accum   <<<9, 256, 0, stream>>>(ws);
    precompute_bw<<<8, 256, 0, stream>>>(Wa_in, ba_in, Wa_out, ba_out, ws);
    stream_softmax<<<dim3(256, 2), 256, 0, stream>>>(X_in, X_out, ws);
    wsum_matvec_wmma<<<128, 32, 0, stream>>>(Ws_in, Ws_out, ws);
    final_linear <<<128, 256, 0, stream>>>(lin_W, lin_b, ws, out);
}